// ParagraphGNN_10685878632941
// MI455X (gfx1250) — compile-verified
//
#include <hip/hip_runtime.h>

// GCN: 3 layers of  x = relu(scatter_add(norm * (x@W)[src] -> dst) + b)
// N=100000 nodes, D=128, E=1.6M edges, symmetric deg^-1/2 normalization w/ self loops.

#define DIM 128

typedef __attribute__((ext_vector_type(2))) float v2f;
typedef __attribute__((ext_vector_type(8))) float v8f;
typedef __attribute__((ext_vector_type(4))) unsigned int u32x4;
typedef __attribute__((ext_vector_type(8))) int i32x8;
typedef __attribute__((ext_vector_type(4))) int i32x4;

#if defined(__has_builtin)
#if __has_builtin(__builtin_amdgcn_tensor_load_to_lds) && __has_builtin(__builtin_amdgcn_s_wait_tensorcnt)
#define USE_TDM 1
#endif
#endif
#ifndef USE_TDM
#define USE_TDM 0
#endif

// ---------------- degree / normalization ----------------

__global__ void k_deg_init(float* __restrict__ deg, int n) {
    int i = blockIdx.x * blockDim.x + threadIdx.x;
    if (i < n) deg[i] = 1.0f;  // self loop
}

__global__ void k_deg_count(const int* __restrict__ dst, float* __restrict__ deg, int e) {
    int i = blockIdx.x * blockDim.x + threadIdx.x;
    if (i < e) atomicAdd(&deg[dst[i]], 1.0f);
}

__global__ void k_deg_rsqrt(float* __restrict__ deg, int n) {
    int i = blockIdx.x * blockDim.x + threadIdx.x;
    if (i < n) deg[i] = rsqrtf(deg[i]);  // deg >= 1 always (self loop)
}

// ---------------- WMMA GEMM:  H[n,128] = X[n,128] @ W[128,128] ----------------
// Persistent blocks (grid-stride over 16-row slabs). 8 waves/block, wave w owns
// cols [16w,16w+16); its full 128x16 B panel is cached in 64 VGPRs for all slabs.
// A slabs (16x128 fp32 = 8KB) are double-buffered in LDS, filled by the Tensor
// Data Mover (TDM): wave 0 issues a 2D tile descriptor, tracks TENSORcnt, and the
// block barriers before compute. fp32 WMMA 16x16x4, K swept in 32 steps.

#if USE_TDM
// Build & issue a D# for a 16x128 f32 tile: global (row-major, stride 128) -> LDS.
__device__ __forceinline__ void tdm_load_tile(const float* gsrc, unsigned ldsOff, int remRows) {
    unsigned long long ga = (unsigned long long)(uintptr_t)gsrc;
    u32x4 g0;
    g0.x = 1u;                                               // count=1, user descriptor
    g0.y = ldsOff;                                           // lds_addr (bytes)
    g0.z = (unsigned)(ga & 0xFFFFFFFFu);                     // global_addr[31:0]
    g0.w = (unsigned)((ga >> 32) & 0x1FFFFFFu) | (2u << 30); // global_addr[56:32] | type=2
    i32x8 g1;
    g1[0] = (int)(2u << 16);                                 // data_size=2 (4 bytes)
    g1[1] = (int)(128u << 16);                               // tensor_dim0 = 128 (lo16)
    g1[2] = (int)(((unsigned)remRows & 0xFFFFu) << 16);      // dim0 hi=0 | tensor_dim1 lo16
    g1[3] = (int)((((unsigned)remRows >> 16) & 0xFFFFu) | (128u << 16)); // dim1 hi | tile_dim0=128
    g1[4] = 16;                                              // tile_dim1=16 | tile_dim2=0
    g1[5] = 128;                                             // tensor_dim0_stride = 128 (lo32)
    g1[6] = 0;                                               // stride0 hi | stride1 lo
    g1[7] = 0;
    i32x4 z4 = {0, 0, 0, 0};
    i32x8 z8 = {0, 0, 0, 0, 0, 0, 0, 0};
    // clang-23 toolchain: 6-arg form (g0, g1, g2, g3, extra, cpol)
    __builtin_amdgcn_tensor_load_to_lds(g0, g1, z4, z4, z8, 0);
}
#endif

__global__ void __launch_bounds__(256)
k_gemm_wmma(const float* __restrict__ X, const float* __restrict__ W,
            float* __restrict__ H, int nrows) {
    __shared__ float ldsA[2][16 * DIM];  // double-buffered A slab (2 x 8KB)

    const int nSlab = (nrows + 15) >> 4;
    const int wave  = threadIdx.x >> 5;
    const int lane  = threadIdx.x & 31;
    const int col0  = wave * 16;
    const int nn    = lane & 15;             // M for A-frag, N for B-frag (ISA 7.12.2)
    const int kh    = (lane >= 16) ? 2 : 0;  // K offset for upper half-wave
    const bool isW0 = (wave == 0);

    // Cache this wave's full B panel (128x16 of W) in registers: 32 steps x v2f.
    v2f breg[32];
#pragma unroll
    for (int kk = 0; kk < 32; ++kk) {
        const int kb = kk * 4 + kh;
        breg[kk].x = W[(size_t)kb * DIM + col0 + nn];
        breg[kk].y = W[(size_t)(kb + 1) * DIM + col0 + nn];
    }

    const int stride = gridDim.x;
    int s0 = blockIdx.x;

#if USE_TDM
    const unsigned ldsBase = (unsigned)(uintptr_t)(void*)&ldsA[0][0];
    if (isW0 && s0 < nSlab)
        tdm_load_tile(X + (size_t)s0 * 16 * DIM, ldsBase, nrows - s0 * 16);
#endif

    int it = 0;
    for (int s = s0; s < nSlab; s += stride, ++it) {
        const int cur = it & 1;
#if USE_TDM
        const int nxt = s + stride;
        if (isW0) {
            if (nxt < nSlab) {
                tdm_load_tile(X + (size_t)nxt * 16 * DIM,
                              ldsBase + (unsigned)((it + 1) & 1) * (16 * DIM * 4),
                              nrows - nxt * 16);
                __builtin_amdgcn_s_wait_tensorcnt(1);  // slab s resident
            } else {
                __builtin_amdgcn_s_wait_tensorcnt(0);
            }
        }
        __syncthreads();  // publish buf[cur] to all waves
#else
        // Fallback: cooperative coalesced staging of the 16x128 slab.
        __syncthreads();
        {
            const int row0 = s * 16;
#pragma unroll
            for (int r = 0; r < 8; ++r) {
                int lin = threadIdx.x + r * 256;
                int row = row0 + (lin >> 7);
                ldsA[cur][lin] = (row < nrows) ? X[(size_t)row * DIM + (lin & 127)] : 0.0f;
            }
        }
        __syncthreads();
#endif

        const float* A = &ldsA[cur][0];
        v8f c = {};
#pragma unroll
        for (int kk = 0; kk < 32; ++kk) {
            const int kb = kk * 4 + kh;
            v2f a;
            a.x = A[nn * DIM + kb];       // A[M=nn, K=kb]   (VGPR0)
            a.y = A[nn * DIM + kb + 1];   // A[M=nn, K=kb+1] (VGPR1)
            c = __builtin_amdgcn_wmma_f32_16x16x4_f32(
                    /*neg_a=*/false, a, /*neg_b=*/false, breg[kk],
                    /*c_mod=*/(short)0, c, /*reuse_a=*/false, /*reuse_b=*/false);
        }

        // C/D layout: VGPR j, lanes 0-15 -> M=j, lanes 16-31 -> M=j+8; N = lane&15.
        const int mbase = s * 16 + ((lane >= 16) ? 8 : 0);
#pragma unroll
        for (int j = 0; j < 8; ++j) {
            int row = mbase + j;
            if (row < nrows) H[(size_t)row * DIM + col0 + nn] = c[j];
        }
        __syncthreads();  // all waves done with buf[cur] before TDM refills it
    }
}

// ---------------- aggregation ----------------

__global__ void k_acc_init(float4* __restrict__ acc, const float4* __restrict__ bias4, int total4) {
    int i = blockIdx.x * blockDim.x + threadIdx.x;
    if (i < total4) acc[i] = bias4[i & 31];  // D/4 = 32 float4 per row
}

// One wave per edge: coalesced float4 gather of h[src], 4 float atomics into acc[dst].
__global__ void __launch_bounds__(256)
k_edge_scatter(const int* __restrict__ src, const int* __restrict__ dst,
               const float* __restrict__ dis, const float* __restrict__ H,
               float* __restrict__ acc, int e) {
    int warp = blockIdx.x * (blockDim.x >> 5) + (threadIdx.x >> 5);
    int lane = threadIdx.x & 31;
    if (warp >= e) return;
    int s = src[warp];
    int d = dst[warp];
    float nrm = dis[s] * dis[d];
    const float4* hp = (const float4*)(H + (size_t)s * DIM);
    float4 v = hp[lane];                       // lanes cover all 128 floats, 512B/wave
    float* ap = acc + (size_t)d * DIM + lane * 4;
    atomicAdd(ap + 0, v.x * nrm);
    atomicAdd(ap + 1, v.y * nrm);
    atomicAdd(ap + 2, v.z * nrm);
    atomicAdd(ap + 3, v.w * nrm);
}

// out = relu(acc + h * dis^2)   (self-loop folded in here), float4-vectorized
__global__ void k_relu_self(const float4* __restrict__ acc, const float4* __restrict__ H,
                            const float* __restrict__ dis, float4* __restrict__ xout, int total4) {
    int i = blockIdx.x * blockDim.x + threadIdx.x;
    if (i < total4) {
        float di = dis[i >> 5];  // node = (i*4)/128
        float n2 = di * di;
        float4 a = acc[i], h = H[i], r;
        r.x = fmaxf(a.x + h.x * n2, 0.0f);
        r.y = fmaxf(a.y + h.y * n2, 0.0f);
        r.z = fmaxf(a.z + h.z * n2, 0.0f);
        r.w = fmaxf(a.w + h.w * n2, 0.0f);
        xout[i] = r;
    }
}

// ---------------- driver ----------------

extern "C" void kernel_launch(void* const* d_in, const int* in_sizes, int n_in,
                              void* d_out, int out_size, void* d_ws, size_t ws_size,
                              hipStream_t stream) {
    const float* x  = (const float*)d_in[0];
    const int*   ei = (const int*)d_in[1];
    const float* W  = (const float*)d_in[2];
    const float* b  = (const float*)d_in[3];

    const int N = in_sizes[0] / DIM;
    const int E = in_sizes[1] / 2;
    const int L = in_sizes[2] / (DIM * DIM);

    const int* srcp = ei;        // edge_index[0]
    const int* dstp = ei + E;    // edge_index[1]

    // Workspace layout (floats): dis[N] | h[N*D] | acc[N*D]
    float* dis = (float*)d_ws;
    size_t nAlign = ((size_t)N + 255) & ~(size_t)255;
    float* h   = dis + nAlign;
    float* acc = h + (size_t)N * DIM;

    float* xout = (float*)d_out;
    const int total  = N * DIM;
    const int total4 = total / 4;

    // Normalization coefficients (computed once).
    k_deg_init <<<(N + 255) / 256, 256, 0, stream>>>(dis, N);
    k_deg_count<<<(E + 255) / 256, 256, 0, stream>>>(dstp, dis, E);
    k_deg_rsqrt<<<(N + 255) / 256, 256, 0, stream>>>(dis, N);

    const int nSlab      = (N + 15) / 16;
    const int gemmBlocks = nSlab < 1250 ? nSlab : 1250;  // persistent, grid-stride
    const int edgeBlocks = (E + 7) / 8;                  // 8 edges (waves) per block
    const int elemBlocks = (total4 + 255) / 256;

    const float* xcur = x;
    for (int l = 0; l < L; ++l) {
        k_gemm_wmma   <<<gemmBlocks, 256, 0, stream>>>(xcur, W + (size_t)l * DIM * DIM, h, N);
        k_acc_init    <<<elemBlocks, 256, 0, stream>>>((float4*)acc, (const float4*)(b + (size_t)l * DIM), total4);
        k_edge_scatter<<<edgeBlocks, 256, 0, stream>>>(srcp, dstp, dis, h, acc, E);
        k_relu_self   <<<elemBlocks, 256, 0, stream>>>((const float4*)acc, (const float4*)h, dis, (float4*)xout, total4);
        xcur = xout;  // next layer reads this layer's output (d_out ping-pongs as x)
    }
}